// LightweightConv_7327214207176
// MI455X (gfx1250) — compile-verified
//
#include <hip/hip_runtime.h>
#include <math.h>

typedef __attribute__((ext_vector_type(2))) float v2f;
typedef __attribute__((ext_vector_type(4))) float v4f;
typedef __attribute__((ext_vector_type(8))) float v8f;
typedef __attribute__((ext_vector_type(4))) int   v4i;

#define B_DIM 8
#define C_DIM 1024
#define T_DIM 2048
#define HEADS 16
#define KTAPS 31
#define CHUNK 256          // outputs per 16x16 WMMA tile
#define NT    (T_DIM / CHUNK)  // 8 tiles per channel, one channel per wave
#define KSTEPS 12          // K padded to 48 = 12 x (K=4)
#define XWIN4 76           // staged window: [t0-16, t0+288) = 304 floats = 76 float4
#define XPADD 320          // LDS floats per buffer (padded)

#if __has_builtin(__builtin_amdgcn_global_load_async_to_lds_b128)
#define HAVE_ASYNC 1
#else
#define HAVE_ASYNC 0
#endif

#if __has_builtin(__builtin_amdgcn_s_wait_asynccnt)
#define WAIT_ASYNC(n) __builtin_amdgcn_s_wait_asynccnt(n)
#else
#define WAIT_ASYNC(n) asm volatile("s_wait_asynccnt %0" ::"i"(n) : "memory")
#endif

// builtin wants: (v4i addrspace(1)* src, v4i addrspace(3)* dst, imm, imm)
#define AS1P(p) ((__attribute__((address_space(1))) v4i*)(void*)(p))
#define AS3P(p) ((__attribute__((address_space(3))) v4i*)(void*)(p))

// ---------------------------------------------------------------------------
// Kernel 0: softmax-normalize (16,1,31) weights into d_ws as [16][32] f32
// ---------------------------------------------------------------------------
__global__ void lconv_softmax_w(const float* __restrict__ w,
                                float* __restrict__ wn) {
    int h = threadIdx.x;
    if (h < HEADS) {
        float mx = -INFINITY;
        for (int k = 0; k < KTAPS; ++k) mx = fmaxf(mx, w[h * KTAPS + k]);
        float s = 0.f;
        for (int k = 0; k < KTAPS; ++k) s += __expf(w[h * KTAPS + k] - mx);
        float inv = 1.f / s;
        for (int k = 0; k < KTAPS; ++k)
            wn[h * 32 + k] = __expf(w[h * KTAPS + k] - mx) * inv;
        wn[h * 32 + 31] = 0.f;
    }
}

// Stage x[t0-16, t0+288) of one channel into a wave-private LDS buffer.
// Window is 64B-aligned; each float4 is entirely in or out of [0, T_DIM).
__device__ __forceinline__ void stage_tile(const float* __restrict__ xrow,
                                           float* buf, int t0, int lane) {
#pragma unroll
    for (int r = 0; r < 3; ++r) {
        int i = lane + 32 * r;                 // float4 index within window
        if (i < XWIN4) {
            int g0 = t0 - 16 + 4 * i;          // global float index (mult of 4)
            bool ok = (g0 >= 0) && (g0 < T_DIM);
#if HAVE_ASYNC
            if (ok) {
                __builtin_amdgcn_global_load_async_to_lds_b128(
                    AS1P(xrow + g0), AS3P(buf + 4 * i), 0, 0);
            } else {
                v4f z = {};
                *(v4f*)(buf + 4 * i) = z;      // zero-fill halo (ds_store_b128)
            }
#else
            v4f v = {};
            if (ok) v = *(const v4f*)(xrow + g0);   // global_load_b128
            *(v4f*)(buf + 4 * i) = v;               // ds_store_b128
#endif
        }
    }
}

// ---------------------------------------------------------------------------
// Kernel 1: 1024 blocks x 256 threads (8 waves). Each wave owns one channel
// and iterates its 8 tiles. Per tile (t0..t0+255):
//   D[m,n] = sum_j X[m,j] * W[j,n],  out[t0 + 16*m + n] = D[m,n]
//   X[m,j] = x[ch, t0 - 16 + 16*m + j]           (A, from LDS-staged x)
//   W[j,n] = w[h, j-1-n] if 0<=j-1-n<31 else 0   (B, weight Toeplitz, regs)
// Double-buffered LDS; tile i+1 staged (async) while tile i computes.
// ---------------------------------------------------------------------------
__global__ __launch_bounds__(256)
void lconv_wmma_kernel(const float* __restrict__ x,
                       const float* __restrict__ wn,
                       float* __restrict__ out) {
    __shared__ __align__(16) float lds[8 * 2 * XPADD];

    const int lane = threadIdx.x & 31;
    const int wv   = threadIdx.x >> 5;
    const int ch   = blockIdx.x * 8 + wv;     // one channel per wave
    const int h    = ch & (HEADS - 1);        // head = c % 16 (C % 16 == 0)

    float* buf0 = lds + wv * (2 * XPADD);
    float* buf1 = buf0 + XPADD;
    const float* __restrict__ xrow = x + (size_t)ch * T_DIM;
    float* __restrict__ obase = out + (size_t)ch * T_DIM;

    // ---- B fragments (weight Toeplitz), built once, held in registers ----
    // B 4x16 layout (mirror of A): lanes 0-15 hold K rows 0,1; 16-31 rows 2,3
    const int n   = lane & 15;
    const int kk0 = (lane >> 4) * 2;
    v2f bfrag[KSTEPS];
#pragma unroll
    for (int j = 0; j < KSTEPS; ++j) {
        int ja = 4 * j + kk0;
        int i0 = ja - 1 - n;      // tap index for K row ja
        int i1 = ja - n;          // tap index for K row ja+1
        v2f b;
        b.x = (i0 >= 0 && i0 < KTAPS) ? wn[h * 32 + i0] : 0.0f;
        b.y = (i1 >= 0 && i1 < KTAPS) ? wn[h * 32 + i1] : 0.0f;
        bfrag[j] = b;
    }

    const int m  = lane & 15;     // A row (coarse time)
    const int hi = lane >> 4;

    stage_tile(xrow, buf0, 0, lane);          // prologue

    for (int tile = 0; tile < NT; ++tile) {
        const int t0 = tile * CHUNK;
        float* cur = (tile & 1) ? buf1 : buf0;
        float* nxt = (tile & 1) ? buf0 : buf1;

        if (tile + 1 < NT) stage_tile(xrow, nxt, t0 + CHUNK, lane);

#if HAVE_ASYNC
        // current tile's 3 async loads retired when <= next tile's 3 remain
        if (tile + 1 < NT) { WAIT_ASYNC(3); } else { WAIT_ASYNC(0); }
#endif

        // ---- 12 chained V_WMMA_F32_16X16X4_F32 ----
        // A 16x4: lane holds row M=lane%16; lanes 0-15 -> K 0,1; 16-31 -> K 2,3
        v8f acc = {};
#pragma unroll
        for (int j = 0; j < KSTEPS; ++j) {
            int ja = 4 * j + kk0;                       // even -> 8B aligned
            v2f a = *(const v2f*)(cur + 16 * m + ja);   // ds_load_b64
            acc = __builtin_amdgcn_wmma_f32_16x16x4_f32(
                false, a, false, bfrag[j], (short)0, acc, false, false);
        }

        // ---- store D: VGPR r -> M = r + 8*hi, N = n; out = t0 + 16*M + N
        float* __restrict__ orow = obase + t0;
#pragma unroll
        for (int r = 0; r < 8; ++r) {
            orow[16 * (r + 8 * hi) + n] = acc[r];
        }
    }
}

extern "C" void kernel_launch(void* const* d_in, const int* in_sizes, int n_in,
                              void* d_out, int out_size, void* d_ws, size_t ws_size,
                              hipStream_t stream) {
    const float* x = (const float*)d_in[0];   // (8,1024,2048) f32
    const float* w = (const float*)d_in[1];   // (16,1,31) f32
    float* out = (float*)d_out;               // (8,1024,2048) f32
    float* wn  = (float*)d_ws;                // [16][32] normalized weights

    lconv_softmax_w<<<1, 32, 0, stream>>>(w, wn);

    const int nblocks = (B_DIM * C_DIM) / 8;  // 1024 blocks, 8 channels each
    lconv_wmma_kernel<<<nblocks, 256, 0, stream>>>(x, wn, out);
}